// GIN_57801669869755
// MI455X (gfx1250) — compile-verified
//
#include <hip/hip_runtime.h>
#include <hip/hip_bf16.h>
#include <math.h>

// Problem constants (from reference)
#define Nn 100000
#define Ee 1600000
#define Ff 128
#define Hh 256
#define Gg 512
#define Cc 10
#define Ll 3
#define BN_EPS 1e-5f

typedef __attribute__((ext_vector_type(2))) float v2f;
typedef __attribute__((ext_vector_type(8))) float v8f;

// -------------------------------------------------------------------------
// zero a float buffer
__global__ void zero_f32(float* __restrict__ p, long long n) {
    long long i = (long long)blockIdx.x * blockDim.x + threadIdx.x;
    if (i < n) p[i] = 0.0f;
}

// -------------------------------------------------------------------------
// Edge-parallel scatter-add: agg[dst[e], :] += h[src[e], :]
// One thread handles a float4 chunk of one edge. cshift = log2(Hcols/4).
__global__ void scatter_add(const float* __restrict__ h,
                            const int* __restrict__ src,
                            const int* __restrict__ dst,
                            float* __restrict__ agg,
                            long long total, int cshift, int Hcols) {
    long long idx = (long long)blockIdx.x * blockDim.x + threadIdx.x;
    if (idx >= total) return;
    int e = (int)(idx >> cshift);
    int c = ((int)idx & ((1 << cshift) - 1)) << 2;
    int s = src[e];
    int d = dst[e];
    const float4 v = *(const float4*)(h + (size_t)s * Hcols + c);
    float* p = agg + (size_t)d * Hcols + c;
    atomicAdd(p + 0, v.x);
    atomicAdd(p + 1, v.y);
    atomicAdd(p + 2, v.z);
    atomicAdd(p + 3, v.w);
}

// -------------------------------------------------------------------------
// Fused GIN GEMM using V_WMMA_F32_16X16X4_F32 (exact fp32 math).
// Out[M x Hout] = epilogue( (A [+ A2]) [M x KK] @ W [KK x Hout] + bias )
// One wave -> 16 rows x 64 cols (4 WMMA accumulators, A fragment reused 4x).
// Block = 8 waves = 128 rows x 64 cols.
//
// W slice staged in LDS with K-PAIR INTERLEAVED layout:
//   wlds[p*128 + 2n + q] = W[2p+q][col0+n]   (q = 0/1)
// so a B fragment {W[ka][n], W[ka+1][n]} (ka always even) is one contiguous
// 8-byte LDS load directly into a register pair -> no VGPR shuffle movs,
// and the 4 sub-tile fragments sit 128 B apart (ds_load_2addr_b64 pairable).
template <bool HAS_A2, bool DO_RELU, bool DO_BN, int KK>
__global__ __launch_bounds__(256) void gemm_wmma_f32(
    const float* __restrict__ A, const float* __restrict__ A2,
    const float* __restrict__ W, const float* __restrict__ bias,
    const float* __restrict__ g, const float* __restrict__ be,
    const float* __restrict__ rm, const float* __restrict__ rv,
    float* __restrict__ Out, int M, int Hout)
{
    __shared__ float wlds[KK * 64];             // (KK/2) pair-rows x 64 cols x 2
    const int col0 = blockIdx.x * 64;           // output column super-tile
    const int lane = threadIdx.x & 31;
    const int wave = threadIdx.x >> 5;

    // cooperative stage: thread handles one (k-pair, n) cell; b64 LDS stores
    for (int i = threadIdx.x; i < (KK / 2) * 64; i += 256) {
        const int p = i >> 6;          // k pair index (k = 2p, 2p+1)
        const int n = i & 63;
        v2f w;
        w[0] = W[(size_t)(2 * p)     * Hout + col0 + n];
        w[1] = W[(size_t)(2 * p + 1) * Hout + col0 + n];
        *(v2f*)&wlds[p * 128 + 2 * n] = w;
    }
    __syncthreads();

    const int row0 = (blockIdx.y * 8 + wave) * 16;   // wave-uniform guard
    if (row0 >= M) return;

    const int m    = lane & 15;            // row within A tile / col within B tile
    const int koff = (lane >> 4) * 2;      // K sub-offset per half-wave (ISA layout)
    const float* arow  = A + (size_t)(row0 + m) * KK;
    const float* arow2 = HAS_A2 ? A2 + (size_t)(row0 + m) * KK : nullptr;
    const float* bptr  = &wlds[(koff >> 1) * 128 + 2 * m];   // per-lane B base

    v8f c0 = {0.f,0.f,0.f,0.f,0.f,0.f,0.f,0.f};
    v8f c1 = c0, c2 = c0, c3 = c0;

    #pragma unroll 4
    for (int k = 0; k < KK; k += 4) {
        const int ka = k + koff;
        v2f a;
        a[0] = arow[ka];
        a[1] = arow[ka + 1];
        if (HAS_A2) { a[0] += arow2[ka]; a[1] += arow2[ka + 1]; }

        const float* bk = bptr + (k >> 1) * 128;   // pair row k/2 (+ lane koff/2)
        v2f b0 = *(const v2f*)(bk +  0);           // n = m
        v2f b1 = *(const v2f*)(bk + 32);           // n = m + 16
        v2f b2 = *(const v2f*)(bk + 64);           // n = m + 32
        v2f b3 = *(const v2f*)(bk + 96);           // n = m + 48

        c0 = __builtin_amdgcn_wmma_f32_16x16x4_f32(false, a, false, b0, (short)0, c0, false, false);
        c1 = __builtin_amdgcn_wmma_f32_16x16x4_f32(false, a, false, b1, (short)0, c1, false, false);
        c2 = __builtin_amdgcn_wmma_f32_16x16x4_f32(false, a, false, b2, (short)0, c2, false, false);
        c3 = __builtin_amdgcn_wmma_f32_16x16x4_f32(false, a, false, b3, (short)0, c3, false, false);
    }

    // epilogue: lane owns 4 columns (one per sub-tile) across 8 rows each
    const int mbase = (lane >> 4) << 3;    // +8 for lanes 16..31
    v8f acc[4] = {c0, c1, c2, c3};
    #pragma unroll
    for (int j = 0; j < 4; ++j) {
        const int n = col0 + j * 16 + m;
        const float bv = bias[n];
        float s = 1.0f, sh = 0.0f;
        if (DO_BN) {
            s  = g[n] * rsqrtf(rv[n] + BN_EPS);
            sh = be[n] - rm[n] * s;
        }
        #pragma unroll
        for (int r = 0; r < 8; ++r) {
            float v = acc[j][r] + bv;
            if (DO_RELU) v = fmaxf(v, 0.0f);
            if (DO_BN)   v = v * s + sh;
            Out[(size_t)(row0 + r + mbase) * Hout + n] = v;
        }
    }
}

// -------------------------------------------------------------------------
// Pooling: per-node atomic accumulation of feature sums and counts
__global__ void pool_sum(const float* __restrict__ h, const int* __restrict__ batch,
                         float* __restrict__ sums) {
    long long idx = (long long)blockIdx.x * blockDim.x + threadIdx.x;
    long long total = (long long)Nn * (Hh / 4);
    if (idx >= total) return;
    int nd = (int)(idx >> 6);
    int c  = ((int)idx & 63) << 2;
    int gi = batch[nd];
    const float4 v = *(const float4*)(h + (size_t)nd * Hh + c);
    float* p = sums + (size_t)gi * Hh + c;
    atomicAdd(p + 0, v.x);
    atomicAdd(p + 1, v.y);
    atomicAdd(p + 2, v.z);
    atomicAdd(p + 3, v.w);
}

__global__ void pool_count(const int* __restrict__ batch, float* __restrict__ counts) {
    int i = blockIdx.x * blockDim.x + threadIdx.x;
    if (i < Nn) atomicAdd(&counts[batch[i]], 1.0f);
}

__global__ void pool_div(float* __restrict__ sums, const float* __restrict__ counts) {
    int i = blockIdx.x * blockDim.x + threadIdx.x;
    if (i < Gg * Hh) sums[i] /= fmaxf(counts[i >> 8], 1.0f);   // Hh == 256
}

// -------------------------------------------------------------------------
// Final tiny GEMM (512 x 256 @ 256 x 10) + bias -> logits
__global__ void head_out(const float* __restrict__ hrelu, const float* __restrict__ Wo,
                         const float* __restrict__ bo, float* __restrict__ out) {
    int i = blockIdx.x * blockDim.x + threadIdx.x;
    if (i >= Gg * Cc) return;
    int gi = i / Cc, ci = i % Cc;
    float acc = bo[ci];
    const float* hr = hrelu + (size_t)gi * Hh;
    for (int k = 0; k < Hh; ++k) acc = fmaf(hr[k], Wo[(size_t)k * Cc + ci], acc);
    out[i] = acc;
}

__global__ void log_softmax_k(float* __restrict__ out) {
    int gi = blockIdx.x * blockDim.x + threadIdx.x;
    if (gi >= Gg) return;
    float* row = out + (size_t)gi * Cc;
    float mx = -INFINITY;
    for (int c = 0; c < Cc; ++c) mx = fmaxf(mx, row[c]);
    float s = 0.0f;
    for (int c = 0; c < Cc; ++c) s += expf(row[c] - mx);
    float l = logf(s) + mx;
    for (int c = 0; c < Cc; ++c) row[c] -= l;
}

// -------------------------------------------------------------------------
template <bool HAS_A2, bool DO_RELU, bool DO_BN, int KK>
static inline void launch_gemm_t(const float* A, const float* A2, const float* W,
                                 const float* bias, const float* g, const float* be,
                                 const float* rm, const float* rv, float* Out,
                                 int M, int Hout, hipStream_t stream) {
    dim3 grid(Hout / 64, (M / 16 + 7) / 8);
    hipLaunchKernelGGL((gemm_wmma_f32<HAS_A2, DO_RELU, DO_BN, KK>),
                       grid, dim3(256), 0, stream,
                       A, A2, W, bias, g, be, rm, rv, Out, M, Hout);
}

extern "C" void kernel_launch(void* const* d_in, const int* in_sizes, int n_in,
                              void* d_out, int out_size, void* d_ws, size_t ws_size,
                              hipStream_t stream) {
    (void)in_sizes; (void)n_in; (void)out_size; (void)ws_size;

    const float* x    = (const float*)d_in[0];
    const int*   eidx = (const int*)  d_in[1];
    const int*   batch= (const int*)  d_in[2];
    const float* W1a  = (const float*)d_in[3];
    const float* b1a  = (const float*)d_in[4];
    const float* W2a  = (const float*)d_in[5];
    const float* b2a  = (const float*)d_in[6];
    const float* g_a  = (const float*)d_in[7];
    const float* be_a = (const float*)d_in[8];
    const float* rm_a = (const float*)d_in[9];
    const float* rv_a = (const float*)d_in[10];
    const float* Ws1  = (const float*)d_in[11];
    const float* bs1  = (const float*)d_in[12];
    const float* Ws2  = (const float*)d_in[13];
    const float* bs2  = (const float*)d_in[14];
    const float* gs   = (const float*)d_in[15];
    const float* bes  = (const float*)d_in[16];
    const float* rms  = (const float*)d_in[17];
    const float* rvs  = (const float*)d_in[18];
    const float* Wf   = (const float*)d_in[19];
    const float* bf   = (const float*)d_in[20];
    const float* Wo   = (const float*)d_in[21];
    const float* bo   = (const float*)d_in[22];
    float* out = (float*)d_out;

    const int* src = eidx;          // edge_index[0]
    const int* dst = eidx + Ee;     // edge_index[1]

    // workspace layout (floats)
    const size_t NH = (size_t)Nn * Hh;
    float* agg    = (float*)d_ws;           // N x H
    float* t      = agg + NH;               // N x H
    float* h      = t + NH;                 // N x H
    float* sums   = h + NH;                 // G x H
    float* counts = sums + (size_t)Gg * Hh; // G
    float* hrelu  = counts + Gg;            // G x H

    const int ZB = 256;
    // ---------------- Layer 1 (F=128 -> H=256) ----------------
    {
        long long nz = (long long)Nn * Ff;
        hipLaunchKernelGGL(zero_f32, dim3((nz + ZB - 1) / ZB), dim3(ZB), 0, stream, agg, nz);
        long long tot = (long long)Ee * (Ff / 4);
        hipLaunchKernelGGL(scatter_add, dim3((tot + ZB - 1) / ZB), dim3(ZB), 0, stream,
                           x, src, dst, agg, tot, 5, Ff);
        // t = relu((x + agg) @ W1a + b1a)
        launch_gemm_t<true, true, false, Ff>(x, agg, W1a, b1a,
                                             nullptr, nullptr, nullptr, nullptr,
                                             t, Nn, Hh, stream);
        // h = BN(relu(t @ W2a + b2a))
        launch_gemm_t<false, true, true, Hh>(t, nullptr, W2a, b2a,
                                             g_a, be_a, rm_a, rv_a,
                                             h, Nn, Hh, stream);
    }

    // ---------------- Layers 2..4 (H -> H) ----------------
    for (int i = 0; i < Ll; ++i) {
        long long nz = (long long)Nn * Hh;
        hipLaunchKernelGGL(zero_f32, dim3((nz + ZB - 1) / ZB), dim3(ZB), 0, stream, agg, nz);
        long long tot = (long long)Ee * (Hh / 4);
        hipLaunchKernelGGL(scatter_add, dim3((tot + ZB - 1) / ZB), dim3(ZB), 0, stream,
                           h, src, dst, agg, tot, 6, Hh);
        launch_gemm_t<true, true, false, Hh>(h, agg,
                                             Ws1 + (size_t)i * Hh * Hh, bs1 + (size_t)i * Hh,
                                             nullptr, nullptr, nullptr, nullptr,
                                             t, Nn, Hh, stream);
        launch_gemm_t<false, true, true, Hh>(t, nullptr,
                                             Ws2 + (size_t)i * Hh * Hh, bs2 + (size_t)i * Hh,
                                             gs + (size_t)i * Hh, bes + (size_t)i * Hh,
                                             rms + (size_t)i * Hh, rvs + (size_t)i * Hh,
                                             h, Nn, Hh, stream);
    }

    // ---------------- Global mean pool ----------------
    {
        long long nz = (long long)Gg * Hh + Gg;   // sums + counts are contiguous
        hipLaunchKernelGGL(zero_f32, dim3((nz + ZB - 1) / ZB), dim3(ZB), 0, stream, sums, nz);
        long long tot = (long long)Nn * (Hh / 4);
        hipLaunchKernelGGL(pool_sum, dim3((tot + ZB - 1) / ZB), dim3(ZB), 0, stream,
                           h, batch, sums);
        hipLaunchKernelGGL(pool_count, dim3((Nn + ZB - 1) / ZB), dim3(ZB), 0, stream,
                           batch, counts);
        hipLaunchKernelGGL(pool_div, dim3((Gg * Hh + ZB - 1) / ZB), dim3(ZB), 0, stream,
                           sums, counts);
    }

    // ---------------- Head ----------------
    // hrelu = relu(pooled @ Wf + bf)  (512 x 256 @ 256 x 256, WMMA)
    launch_gemm_t<false, true, false, Hh>(sums, nullptr, Wf, bf,
                                          nullptr, nullptr, nullptr, nullptr,
                                          hrelu, Gg, Hh, stream);
    // logits + log_softmax
    hipLaunchKernelGGL(head_out, dim3((Gg * Cc + ZB - 1) / ZB), dim3(ZB), 0, stream,
                       hrelu, Wo, bo, out);
    hipLaunchKernelGGL(log_softmax_k, dim3((Gg + ZB - 1) / ZB), dim3(ZB), 0, stream, out);
}